// GraphSAGE_8718783611326
// MI455X (gfx1250) — compile-verified
//
#include <hip/hip_runtime.h>
#include <hip/hip_bf16.h>

// GraphSAGE forward for MI455X (gfx1250, wave32).
//   layer1: mean-aggregate(x) @ W1 + b1 -> relu -> h [N,128]
//   layer2: mean-aggregate(h) @ W2 + b2 -> log_softmax -> out [N,64]
// Scatter (segment_sum) via fp32 global atomics (working set fits 192MB L2);
// GEMMs via V_WMMA_F32_16X16X4_F32 with fused epilogues.

#define GNODES 50000
#define GEPS   1e-6f

typedef __attribute__((ext_vector_type(2))) float v2f;
typedef __attribute__((ext_vector_type(8))) float v8f;

static __device__ __forceinline__ v8f wmma_f32(v2f a, v2f b, v8f c) {
  // (neg_a, A, neg_b, B, c_mod, C, reuse_a, reuse_b)
  return __builtin_amdgcn_wmma_f32_16x16x4_f32(false, a, false, b, (short)0, c,
                                               false, false);
}

// ---------------------------------------------------------------------------
// scratch zeroing (float4 grid)
// ---------------------------------------------------------------------------
__global__ void zero_f4(float4* __restrict__ p, int n4) {
  int i = blockIdx.x * blockDim.x + threadIdx.x;
  if (i < n4) p[i] = make_float4(0.f, 0.f, 0.f, 0.f);
}

// ---------------------------------------------------------------------------
// Edge scatter, layer 1: num[dst] += x[src] (64 feats), deg[dst] += 1.
// One wave per edge; lane covers feats {lane, lane+32}. Row gathers are two
// coalesced 128B requests; adds land in L2.
// ---------------------------------------------------------------------------
__global__ void edge_agg_64(const float* __restrict__ x,
                            const int* __restrict__ src,
                            const int* __restrict__ dst,
                            float* __restrict__ num,
                            float* __restrict__ deg, int E) {
  int wid = (blockIdx.x * blockDim.x + threadIdx.x) >> 5;  // wave-uniform
  if (wid >= E) return;
  int lane = threadIdx.x & 31;
  int s = src[wid];
  int d = dst[wid];
  const float* xr = x + (size_t)s * 64;
  float* nr = num + (size_t)d * 64;
  float v0 = xr[lane];
  float v1 = xr[lane + 32];
  atomicAdd(&nr[lane], v0);
  atomicAdd(&nr[lane + 32], v1);
  if (lane == 0) atomicAdd(&deg[d], 1.0f);
}

// Edge scatter, layer 2: num[dst] += h[src] (128 feats).
__global__ void edge_agg_128(const float* __restrict__ h,
                             const int* __restrict__ src,
                             const int* __restrict__ dst,
                             float* __restrict__ num, int E) {
  int wid = (blockIdx.x * blockDim.x + threadIdx.x) >> 5;
  if (wid >= E) return;
  int lane = threadIdx.x & 31;
  int s = src[wid];
  int d = dst[wid];
  const float* hr = h + (size_t)s * 128;
  float* nr = num + (size_t)d * 128;
#pragma unroll
  for (int j = 0; j < 4; ++j) {
    int f = lane + 32 * j;
    atomicAdd(&nr[f], hr[f]);
  }
}

// ---------------------------------------------------------------------------
// GEMM1: h = relu( (num/deg) @ W1 + b1 ),  [N,64] x [64,128] -> [N,128]
// grid = N/16 blocks of 256 threads; wave w handles column tile w (16 cols).
// 16 chained V_WMMA_F32_16X16X4_F32 per tile (K = 64).
// ---------------------------------------------------------------------------
__global__ void gemm1_relu(const float* __restrict__ num,
                           const float* __restrict__ deg,
                           const float* __restrict__ W1,
                           const float* __restrict__ b1,
                           float* __restrict__ h) {
  const int rowTile = blockIdx.x;           // 0..3124
  const int colTile = threadIdx.x >> 5;     // 0..7
  const int lane    = threadIdx.x & 31;
  const int l15     = lane & 15;
  const int m       = rowTile * 16 + l15;           // A-row for this lane
  const int khalf   = (lane >> 4) << 1;             // 0 or 2
  const int n       = colTile * 16 + l15;           // B/C column

  const float rdeg = 1.0f / (deg[m] + GEPS);
  const float* arow = num + (size_t)m * 64;

  v8f c = {};
#pragma unroll
  for (int k = 0; k < 64; k += 4) {
    const int kk = k + khalf;
    v2f a, b;
    a[0] = arow[kk] * rdeg;
    a[1] = arow[kk + 1] * rdeg;
    b[0] = W1[(size_t)kk * 128 + n];
    b[1] = W1[(size_t)(kk + 1) * 128 + n];
    c = wmma_f32(a, b, c);
  }

  const float bv = b1[n];
  const int rbase = rowTile * 16 + ((lane >> 4) << 3);  // rows i or 8+i
#pragma unroll
  for (int i = 0; i < 8; ++i) {
    float v = c[i] + bv;
    h[(size_t)(rbase + i) * 128 + n] = fmaxf(v, 0.0f);
  }
}

// ---------------------------------------------------------------------------
// GEMM2 + log_softmax: out = log_softmax( (num/deg) @ W2 + b2 ), [N,128]x[128,64]
// One wave per 16-row stripe; 4 accumulators cover the 64 output cols.
// Row-wise max / logsumexp via 16-lane __shfl_xor (masks 1,2,4,8 stay within
// each half of the wave32, matching the C-matrix row layout).
// ---------------------------------------------------------------------------
__global__ void gemm2_logsoftmax(const float* __restrict__ num,
                                 const float* __restrict__ deg,
                                 const float* __restrict__ W2,
                                 const float* __restrict__ b2,
                                 float* __restrict__ out) {
  const int rowTile = blockIdx.x;           // blockDim = 32 (one wave)
  const int lane    = threadIdx.x & 31;
  const int nl      = lane & 15;
  const int m       = rowTile * 16 + nl;
  const int khalf   = (lane >> 4) << 1;

  const float rdeg = 1.0f / (deg[m] + GEPS);
  const float* arow = num + (size_t)m * 128;

  v8f c0 = {}, c1 = {}, c2 = {}, c3 = {};
  for (int k = 0; k < 128; k += 4) {
    const int kk = k + khalf;
    v2f a;
    a[0] = arow[kk] * rdeg;
    a[1] = arow[kk + 1] * rdeg;
    const float* w0 = W2 + (size_t)kk * 64;
    const float* w1 = W2 + (size_t)(kk + 1) * 64;
    v2f b;
    b[0] = w0[nl];      b[1] = w1[nl];      c0 = wmma_f32(a, b, c0);
    b[0] = w0[nl + 16]; b[1] = w1[nl + 16]; c1 = wmma_f32(a, b, c1);
    b[0] = w0[nl + 32]; b[1] = w1[nl + 32]; c2 = wmma_f32(a, b, c2);
    b[0] = w0[nl + 48]; b[1] = w1[nl + 48]; c3 = wmma_f32(a, b, c3);
  }

  const float bv0 = b2[nl], bv1 = b2[nl + 16], bv2 = b2[nl + 32], bv3 = b2[nl + 48];
  const int rbase = rowTile * 16 + ((lane >> 4) << 3);

#pragma unroll
  for (int i = 0; i < 8; ++i) {
    float v0 = c0[i] + bv0, v1 = c1[i] + bv1, v2 = c2[i] + bv2, v3 = c3[i] + bv3;

    float mx = fmaxf(fmaxf(v0, v1), fmaxf(v2, v3));
#pragma unroll
    for (int off = 1; off <= 8; off <<= 1)
      mx = fmaxf(mx, __shfl_xor(mx, off, 32));

    float s = expf(v0 - mx) + expf(v1 - mx) + expf(v2 - mx) + expf(v3 - mx);
#pragma unroll
    for (int off = 1; off <= 8; off <<= 1)
      s += __shfl_xor(s, off, 32);

    const float lse = mx + logf(s);
    float* orow = out + (size_t)(rbase + i) * 64;
    orow[nl]      = v0 - lse;
    orow[nl + 16] = v1 - lse;
    orow[nl + 32] = v2 - lse;
    orow[nl + 48] = v3 - lse;
  }
}

// ---------------------------------------------------------------------------
// launch
// ---------------------------------------------------------------------------
extern "C" void kernel_launch(void* const* d_in, const int* in_sizes, int n_in,
                              void* d_out, int out_size, void* d_ws, size_t ws_size,
                              hipStream_t stream) {
  (void)n_in; (void)out_size; (void)ws_size;

  const float* x   = (const float*)d_in[0];
  const int*   src = (const int*)d_in[1];
  const int*   dst = (const int*)d_in[2];
  const float* W1  = (const float*)d_in[3];
  const float* b1  = (const float*)d_in[4];
  const float* W2  = (const float*)d_in[5];
  const float* b2  = (const float*)d_in[6];
  float*       out = (float*)d_out;

  const int N = GNODES;
  const int E = in_sizes[1];

  // workspace layout (floats):
  //   deg : [0, N)
  //   num : [N, N + N*128)          (layer1 uses first N*64 with stride 64)
  //   h   : [N + N*128, N + 2*N*128)
  float* ws  = (float*)d_ws;
  float* deg = ws;
  float* num = ws + N;
  float* h   = ws + N + (size_t)N * 128;

  // 1) zero deg + num1 region
  {
    int n4 = (N + N * 64) / 4;
    zero_f4<<<(n4 + 255) / 256, 256, 0, stream>>>((float4*)ws, n4);
  }
  // 2) layer-1 scatter: num1 += x[src], deg += 1
  edge_agg_64<<<(E + 7) / 8, 256, 0, stream>>>(x, src, dst, num, deg, E);
  // 3) h = relu(mean1 @ W1 + b1)
  gemm1_relu<<<N / 16, 256, 0, stream>>>(num, deg, W1, b1, h);
  // 4) zero num2 region
  {
    int n4 = (N * 128) / 4;
    zero_f4<<<(n4 + 255) / 256, 256, 0, stream>>>((float4*)num, n4);
  }
  // 5) layer-2 scatter: num2 += h[src]
  edge_agg_128<<<(E + 7) / 8, 256, 0, stream>>>(h, src, dst, num, E);
  // 6) out = log_softmax(mean2 @ W2 + b2)
  gemm2_logsoftmax<<<N / 16, 32, 0, stream>>>(num, deg, W2, b2, out);
}